// Cy2C_GCN_OGB_VN_attr_30039001268363
// MI455X (gfx1250) — compile-verified
//
#include <hip/hip_runtime.h>
#include <hip/hip_bf16.h>

typedef __attribute__((ext_vector_type(2))) float v2f;
typedef __attribute__((ext_vector_type(8))) float v8f;

#define HD 64
#define EPS 1e-5f

// ---------------- encoders ----------------

__global__ __launch_bounds__(256) void k_atom_enc(const int* __restrict__ x,
    const float* __restrict__ atom_emb, float* __restrict__ x0, int N) {
  int idx = blockIdx.x * 256 + threadIdx.x;
  if (idx >= N * HD) return;
  int n = idx >> 6, h = idx & 63;
  float s = 0.f;
#pragma unroll
  for (int j = 0; j < 9; ++j) {
    int v = x[n * 9 + j];
    s += atom_emb[((size_t)(j * 119 + v)) * HD + h];
  }
  x0[idx] = s;
}

__global__ __launch_bounds__(256) void k_bond_enc(const int* __restrict__ edge_attr,
    const float* __restrict__ bond_emb, float* __restrict__ ew, int E) {
  int e = blockIdx.x * 256 + threadIdx.x;
  if (e >= E) return;
  float s = 0.f;
#pragma unroll
  for (int j = 0; j < 3; ++j) s += bond_emb[j * 6 + edge_attr[e * 3 + j]];
  ew[e] = s;
}

// ---------------- WMMA fp32 GEMM: Y[M,64] = X[M,64] @ W[64,64] (+ bias) ----------------
// one wave -> 16x64 output strip; 8 waves/block -> 128 rows/block.
// A 16x4 f32 frag: lane l holds row (l&15), K = 2*(l>>4) + {0,1}
// B 4x16  f32 frag: lane l holds col (l&15), K = 2*(l>>4) + {0,1}
// D 16x16 f32: vgpr r, lane l -> row r + 8*(l>>4), col (l&15)

__global__ __launch_bounds__(256) void k_gemm64(const float* __restrict__ X,
    const float* __restrict__ W, const float* __restrict__ bias,
    float* __restrict__ Y, int M) {
  __shared__ float sW[HD * HD];
  int tid = threadIdx.x;
  for (int i = tid; i < HD * HD; i += 256) sW[i] = W[i];
  __syncthreads();

  int lane = tid & 31;
  int wave = tid >> 5;
  int rowBase = blockIdx.x * 128 + wave * 16;
  int mrow = lane & 15;
  int khalf = (lane >> 4) << 1;   // 0 or 2

  int row = rowBase + mrow;
  bool inb = row < M;
  const float* xrow = X + (size_t)(inb ? row : 0) * HD;

  v2f a[16];
#pragma unroll
  for (int k = 0; k < 16; ++k) {
    a[k].x = inb ? xrow[4 * k + khalf + 0] : 0.f;
    a[k].y = inb ? xrow[4 * k + khalf + 1] : 0.f;
  }

#pragma unroll
  for (int n = 0; n < 4; ++n) {
    v8f c = {};
#pragma unroll
    for (int k = 0; k < 16; ++k) {
      v2f b;
      b.x = sW[(4 * k + khalf + 0) * HD + n * 16 + mrow];
      b.y = sW[(4 * k + khalf + 1) * HD + n * 16 + mrow];
      c = __builtin_amdgcn_wmma_f32_16x16x4_f32(false, a[k], false, b,
                                                (short)0, c, false, false);
    }
    float bc = bias ? bias[n * 16 + mrow] : 0.f;
#pragma unroll
    for (int r = 0; r < 8; ++r) {
      int orow = rowBase + r + ((lane >> 4) << 3);
      if (orow < M) Y[(size_t)orow * HD + n * 16 + mrow] = c[r] + bc;
    }
  }
}

// ---------------- misc elementwise / scatter ----------------

__global__ __launch_bounds__(256) void k_fill_bias(float* __restrict__ buf,
    const float* __restrict__ bias, int M) {
  int idx = blockIdx.x * 256 + threadIdx.x;
  if (idx < M * HD) buf[idx] = bias[idx & 63];
}

__global__ __launch_bounds__(256) void k_vn_init(float* __restrict__ VN0,
    const float* __restrict__ vn_w, int GH) {
  int idx = blockIdx.x * 256 + threadIdx.x;
  if (idx < GH) VN0[idx] = vn_w[idx & 63];
}

__global__ __launch_bounds__(256) void k_add_vn(float* __restrict__ h,
    const float* __restrict__ VN0, const int* __restrict__ batch, int N) {
  int idx = blockIdx.x * 256 + threadIdx.x;
  if (idx >= N * HD) return;
  int n = idx >> 6, hh = idx & 63;
  h[idx] += VN0[(size_t)batch[n] * HD + hh];
}

__global__ __launch_bounds__(256) void k_edge_scatter(const float* __restrict__ hw,
    const float* __restrict__ ew, const int* __restrict__ src,
    const int* __restrict__ dst, float* __restrict__ agg, int E) {
  int idx = blockIdx.x * 256 + threadIdx.x;
  if (idx >= E * HD) return;
  int e = idx >> 6, h = idx & 63;
  int s = src[e], d = dst[e];
  atomicAdd(&agg[(size_t)d * HD + h], ew[e] * hw[(size_t)s * HD + h]);
}

// ---------------- batchnorm (training-mode batch stats) ----------------

__global__ __launch_bounds__(256) void k_bn_stats(const float* __restrict__ xin,
    float* __restrict__ stats, int M) {
  __shared__ float ss[256], sq[256];
  int tid = threadIdx.x;
  int h = tid & 63;
  float s = 0.f, q = 0.f;
  for (int r = blockIdx.x * 4 + (tid >> 6); r < M; r += gridDim.x * 4) {
    float v = xin[(size_t)r * HD + h];
    s += v; q += v * v;
  }
  ss[tid] = s; sq[tid] = q;
  __syncthreads();
  if (tid < 64) {
    s = ss[tid] + ss[tid + 64] + ss[tid + 128] + ss[tid + 192];
    q = sq[tid] + sq[tid + 64] + sq[tid + 128] + sq[tid + 192];
    atomicAdd(&stats[h], s);
    atomicAdd(&stats[64 + h], q);
  }
}

// out = relu(g*(x-mu)*rsqrt(var+eps)+b) (+ resid)
__global__ __launch_bounds__(256) void k_bn_apply(const float* __restrict__ xin,
    const float* __restrict__ stats, const float* __restrict__ gamma,
    const float* __restrict__ beta, const float* __restrict__ resid,
    float* __restrict__ out, int M) {
  int idx = blockIdx.x * 256 + threadIdx.x;
  if (idx >= M * HD) return;
  int h = idx & 63;
  float inv = 1.f / (float)M;
  float mu = stats[h] * inv;
  float var = stats[64 + h] * inv - mu * mu;
  float v = gamma[h] * (xin[idx] - mu) * rsqrtf(var + EPS) + beta[h];
  v = fmaxf(v, 0.f);
  if (resid) v += resid[idx];
  out[idx] = v;
}

// ---------------- VN segment sum ----------------

__global__ __launch_bounds__(256) void k_vn_seg(const float* __restrict__ h,
    const int* __restrict__ batch, float* __restrict__ vg, int N) {
  int idx = blockIdx.x * 256 + threadIdx.x;
  if (idx >= N * HD) return;
  int n = idx >> 6, hh = idx & 63;
  atomicAdd(&vg[(size_t)batch[n] * HD + hh], h[idx]);
}

// ---------------- cycle branch ----------------

__global__ __launch_bounds__(256) void k_deg_init(float* __restrict__ deg, int N) {
  int n = blockIdx.x * 256 + threadIdx.x;
  if (n < N) deg[n] = 1.f;  // self-loop
}

__global__ __launch_bounds__(256) void k_deg_atomic(const int* __restrict__ cdst,
    float* __restrict__ deg, int EC) {
  int e = blockIdx.x * 256 + threadIdx.x;
  if (e < EC) atomicAdd(&deg[cdst[e]], 1.f);
}

__global__ __launch_bounds__(256) void k_deg_fin(float* __restrict__ deg, int N) {
  int n = blockIdx.x * 256 + threadIdx.x;
  if (n < N) deg[n] = rsqrtf(deg[n]);
}

// agg = conv2B + dinv[n]^2 * xw[n]   (self-loop term doubles as initializer)
__global__ __launch_bounds__(256) void k_cycle_init(float* __restrict__ agg,
    const float* __restrict__ conv2B, const float* __restrict__ dinv,
    const float* __restrict__ xw, int N) {
  int idx = blockIdx.x * 256 + threadIdx.x;
  if (idx >= N * HD) return;
  int n = idx >> 6, h = idx & 63;
  float di = dinv[n];
  agg[idx] = conv2B[h] + di * di * xw[idx];
}

__global__ __launch_bounds__(256) void k_cycle_scatter(const float* __restrict__ xw,
    const float* __restrict__ dinv, const int* __restrict__ csrc,
    const int* __restrict__ cdst, float* __restrict__ agg, int EC) {
  int idx = blockIdx.x * 256 + threadIdx.x;
  if (idx >= EC * HD) return;
  int e = idx >> 6, h = idx & 63;
  int s = csrc[e], d = cdst[e];
  atomicAdd(&agg[(size_t)d * HD + h], dinv[s] * dinv[d] * xw[(size_t)s * HD + h]);
}

// ---------------- pooling + final linear ----------------

__global__ __launch_bounds__(256) void k_pool_scatter(const float* __restrict__ xa,
    const float* __restrict__ xb, const int* __restrict__ batch,
    float* __restrict__ poolA, float* __restrict__ poolB,
    float* __restrict__ cnt, int N) {
  int idx = blockIdx.x * 256 + threadIdx.x;
  if (idx >= N * HD) return;
  int n = idx >> 6, h = idx & 63;
  int g = batch[n];
  atomicAdd(&poolA[(size_t)g * HD + h], xa[idx]);
  atomicAdd(&poolB[(size_t)g * HD + h], xb[idx]);
  if (h == 0) atomicAdd(&cnt[g], 1.f);
}

__global__ __launch_bounds__(256) void k_final(const float* __restrict__ poolA,
    const float* __restrict__ poolB, const float* __restrict__ cnt,
    const float* __restrict__ lin1W, const float* __restrict__ lin1b,
    float* __restrict__ out, int G) {
  int g = blockIdx.x * 256 + threadIdx.x;
  if (g >= G) return;
  float c = fmaxf(cnt[g], 1.f);
  float inv = 1.f / c;
  float acc = lin1b[0];
#pragma unroll 8
  for (int h = 0; h < HD; ++h)
    acc += poolA[(size_t)g * HD + h] * inv * lin1W[h]
         + poolB[(size_t)g * HD + h] * inv * lin1W[HD + h];
  out[g] = acc;
}

// ---------------- host orchestration ----------------

extern "C" void kernel_launch(void* const* d_in, const int* in_sizes, int n_in,
                              void* d_out, int out_size, void* d_ws, size_t ws_size,
                              hipStream_t stream) {
  const int*   x          = (const int*)d_in[0];
  const int*   edge_index = (const int*)d_in[1];
  const int*   cycle_index= (const int*)d_in[2];
  const int*   batch      = (const int*)d_in[3];
  const int*   edge_attr  = (const int*)d_in[4];
  const float* atom_emb   = (const float*)d_in[5];
  const float* bond_emb   = (const float*)d_in[6];
  const float* convW      = (const float*)d_in[7];
  const float* convB      = (const float*)d_in[8];
  const float* bnG        = (const float*)d_in[9];
  const float* bnB        = (const float*)d_in[10];
  const float* conv2W     = (const float*)d_in[11];
  const float* conv2B     = (const float*)d_in[12];
  const float* bn2G       = (const float*)d_in[13];
  const float* bn2B       = (const float*)d_in[14];
  const float* mlpW1      = (const float*)d_in[15];
  const float* mlpB1      = (const float*)d_in[16];
  const float* mlpG1      = (const float*)d_in[17];
  const float* mlpBe1     = (const float*)d_in[18];
  const float* mlpW2      = (const float*)d_in[19];
  const float* mlpB2      = (const float*)d_in[20];
  const float* mlpG2      = (const float*)d_in[21];
  const float* mlpBe2     = (const float*)d_in[22];
  const float* vn_w       = (const float*)d_in[23];
  const float* lin1W      = (const float*)d_in[24];
  const float* lin1b      = (const float*)d_in[25];
  (void)n_in; (void)ws_size;

  const int N  = in_sizes[3];
  const int E  = in_sizes[1] / 2;
  const int EC = in_sizes[2] / 2;
  const int G  = out_size;          // OUT == 1

  const int* src  = edge_index;
  const int* dst  = edge_index + E;
  const int* csrc = cycle_index;
  const int* cdst = cycle_index + EC;

  float* ws = (float*)d_ws;
  size_t off = 0;
  auto alloc = [&](size_t nelem) {
    float* p = ws + off; off += (nelem + 63) & ~(size_t)63; return p;
  };
  float* x0    = alloc((size_t)N * HD);
  float* h     = alloc((size_t)N * HD);
  float* t1    = alloc((size_t)N * HD);   // hw / xw / x_out_b
  float* t2    = alloc((size_t)N * HD);   // agg
  float* ew    = alloc((size_t)E);
  float* dinv  = alloc((size_t)N);
  float* VN0   = alloc((size_t)G * HD);
  float* vg1   = alloc((size_t)G * HD);
  float* vg2   = alloc((size_t)G * HD);
  float* poolA = alloc((size_t)G * HD);
  float* poolB = alloc((size_t)G * HD);
  float* cnt   = alloc((size_t)G);
  float* stats = alloc(128);

  auto cdiv = [](long long a, long long b) { return (unsigned)((a + b - 1) / b); };
  const unsigned gNH  = cdiv((long long)N * HD, 256);
  const unsigned gGH  = cdiv((long long)G * HD, 256);
  const unsigned gEH  = cdiv((long long)E * HD, 256);
  const unsigned gECH = cdiv((long long)EC * HD, 256);
  const unsigned gGemmN = cdiv(N, 128);
  const unsigned gGemmG = cdiv(G, 128);

  // encoders
  k_atom_enc<<<gNH, 256, 0, stream>>>(x, atom_emb, x0, N);
  k_bond_enc<<<cdiv(E, 256), 256, 0, stream>>>(edge_attr, bond_emb, ew, E);

  // h = x0 ; VN0 = broadcast(vn_w)
  hipMemcpyAsync(h, x0, (size_t)N * HD * sizeof(float), hipMemcpyDeviceToDevice, stream);
  k_vn_init<<<gGH, 256, 0, stream>>>(VN0, vn_w, G * HD);

  for (int i = 0; i < 3; ++i) {
    k_add_vn<<<gNH, 256, 0, stream>>>(h, VN0, batch, N);
    k_gemm64<<<gGemmN, 256, 0, stream>>>(h, convW + (size_t)i * HD * HD, nullptr, t1, N);
    k_fill_bias<<<gNH, 256, 0, stream>>>(t2, convB + (size_t)i * HD, N);
    k_edge_scatter<<<gEH, 256, 0, stream>>>(t1, ew, src, dst, t2, E);
    hipMemsetAsync(stats, 0, 128 * sizeof(float), stream);
    k_bn_stats<<<512, 256, 0, stream>>>(t2, stats, N);
    k_bn_apply<<<gNH, 256, 0, stream>>>(t2, stats, bnG + (size_t)i * HD,
                                        bnB + (size_t)i * HD, h, h, N);
    if (i < 2) {
      // vna = segment_sum(h, batch) + VN0
      hipMemcpyAsync(vg1, VN0, (size_t)G * HD * sizeof(float), hipMemcpyDeviceToDevice, stream);
      k_vn_seg<<<gNH, 256, 0, stream>>>(h, batch, vg1, N);
      // t = relu(bn(vna @ W1 + b1))
      k_gemm64<<<gGemmG, 256, 0, stream>>>(vg1, mlpW1 + (size_t)i * HD * HD,
                                           mlpB1 + (size_t)i * HD, vg2, G);
      hipMemsetAsync(stats, 0, 128 * sizeof(float), stream);
      k_bn_stats<<<512, 256, 0, stream>>>(vg2, stats, G);
      k_bn_apply<<<gGH, 256, 0, stream>>>(vg2, stats, mlpG1 + (size_t)i * HD,
                                          mlpBe1 + (size_t)i * HD, nullptr, vg1, G);
      // t = relu(bn(t @ W2 + b2)); VN0 += t
      k_gemm64<<<gGemmG, 256, 0, stream>>>(vg1, mlpW2 + (size_t)i * HD * HD,
                                           mlpB2 + (size_t)i * HD, vg2, G);
      hipMemsetAsync(stats, 0, 128 * sizeof(float), stream);
      k_bn_stats<<<512, 256, 0, stream>>>(vg2, stats, G);
      k_bn_apply<<<gGH, 256, 0, stream>>>(vg2, stats, mlpG2 + (size_t)i * HD,
                                          mlpBe2 + (size_t)i * HD, VN0, VN0, G);
    }
  }

  // cycle branch
  k_deg_init<<<cdiv(N, 256), 256, 0, stream>>>(dinv, N);
  k_deg_atomic<<<cdiv(EC, 256), 256, 0, stream>>>(cdst, dinv, EC);
  k_deg_fin<<<cdiv(N, 256), 256, 0, stream>>>(dinv, N);
  k_gemm64<<<gGemmN, 256, 0, stream>>>(x0, conv2W, nullptr, t1, N);
  k_cycle_init<<<gNH, 256, 0, stream>>>(t2, conv2B, dinv, t1, N);
  k_cycle_scatter<<<gECH, 256, 0, stream>>>(t1, dinv, csrc, cdst, t2, EC);
  hipMemsetAsync(stats, 0, 128 * sizeof(float), stream);
  k_bn_stats<<<512, 256, 0, stream>>>(t2, stats, N);
  k_bn_apply<<<gNH, 256, 0, stream>>>(t2, stats, bn2G, bn2B, x0, t1, N);  // t1 = x_out_b

  // pooling + final linear
  hipMemsetAsync(poolA, 0, (size_t)G * HD * sizeof(float), stream);
  hipMemsetAsync(poolB, 0, (size_t)G * HD * sizeof(float), stream);
  hipMemsetAsync(cnt, 0, (size_t)G * sizeof(float), stream);
  k_pool_scatter<<<gNH, 256, 0, stream>>>(h, t1, batch, poolA, poolB, cnt, N);
  k_final<<<cdiv(G, 256), 256, 0, stream>>>(poolA, poolB, cnt, lin1W, lin1b,
                                            (float*)d_out, G);
}